// PSDNorm_62354335203943
// MI455X (gfx1250) — compile-verified
//
#include <hip/hip_runtime.h>
#include <hip/hip_bf16.h>

// ---------------- problem constants ----------------
#define NB   512
#define NC   8
#define NT   6000
#define NK   256          // filter length / nperseg
#define NF   129          // rfft bins
#define NFP  144          // freq bins padded to WMMA tile multiple (zeros beyond 128)
#define NSEG 45           // (6000-128)/128
#define NBC  (NB*NC)      // 4096
#define TWO_PI 6.28318530717958647692f

typedef float v2f __attribute__((ext_vector_type(2)));
typedef float v8f __attribute__((ext_vector_type(8)));
typedef unsigned int u32x4 __attribute__((ext_vector_type(4)));
typedef int i32x4 __attribute__((ext_vector_type(4)));
typedef int i32x8 __attribute__((ext_vector_type(8)));

// fp32 WMMA: D(16x16) = A(16x4) * B(4x16) + C   (keeps reference fp32 precision)
static __device__ inline v8f wmma4(v2f a, v2f b, v8f c) {
    return __builtin_amdgcn_wmma_f32_16x16x4_f32(
        /*neg_a=*/false, a, /*neg_b=*/false, b,
        /*c_mod=*/(short)0, c, /*reuse_a=*/false, /*reuse_b=*/false);
}

// ---------------- Tensor Data Mover: 2D strided tile -> LDS ----------------
// D# group0: count=1 | lds_addr | 57b global_addr | type=2
// D# group1: data_size=4B, pad_enable, pad_interval (2^(n+1) dwords), pad_amount
//            (n+1 dwords), tensor_dim0/1 (OOB rows/cols zero-filled), tile_dim0/1,
//            tensor_dim0_stride.
static __device__ inline void tdm_load_tile(unsigned lds_byte, unsigned long long gbyte,
                                            unsigned dim0, unsigned dim1,
                                            unsigned tile0, unsigned tile1,
                                            unsigned stride0,
                                            unsigned pad_interval, unsigned pad_amount,
                                            bool pad_en) {
    u32x4 g0;
    g0.x = 1u;                                                    // count=1
    g0.y = lds_byte;
    g0.z = (unsigned)(gbyte & 0xffffffffull);
    g0.w = (unsigned)((gbyte >> 32) & 0x1ffffffull) | (2u << 30); // type=2 ("image")
    i32x8 g1;
    g1[0] = (int)((2u << 16) | ((pad_en ? 1u : 0u) << 20)
                  | (pad_interval << 22) | (pad_amount << 25));
    g1[1] = (int)(dim0 << 16);                    // tensor_dim0[15:0]
    g1[2] = (int)((dim0 >> 16) | (dim1 << 16));   // tensor_dim0[31:16] | tensor_dim1[15:0]
    g1[3] = (int)((dim1 >> 16) | (tile0 << 16));  // tensor_dim1[31:16] | tile_dim0
    g1[4] = (int)tile1;                           // tile_dim1 (tile_dim2 = 0)
    g1[5] = (int)stride0;                         // tensor_dim0_stride[31:0]
    g1[6] = 0;
    g1[7] = 0;
    i32x4 z4 = {0, 0, 0, 0};
#if __clang_major__ >= 23
    i32x8 z8 = {0, 0, 0, 0, 0, 0, 0, 0};
    __builtin_amdgcn_tensor_load_to_lds(g0, g1, z4, z4, z8, 0);
#else
    __builtin_amdgcn_tensor_load_to_lds(g0, g1, z4, z4, 0);
#endif
}

// ---------------- K0: DFT / irfft tables ----------------
__global__ void k_init(float* __restrict__ wcp, float* __restrict__ wsp,
                       float* __restrict__ tir) {
    int idx = blockIdx.x * blockDim.x + threadIdx.x;
    if (idx < 128 * NFP) {
        int pair = idx / NFP, f = idx % NFP;
        int n = 2 * pair;
        float c0 = 0.f, c1 = 0.f, s0 = 0.f, s1 = 0.f;
        if (f < NF) {
            float a0 = TWO_PI * (float)f * (float)n       / 256.0f;
            float a1 = TWO_PI * (float)f * (float)(n + 1) / 256.0f;
            c0 = __cosf(a0); c1 = __cosf(a1);
            s0 = __sinf(a0); s1 = __sinf(a1);
        }
        wcp[pair * (2 * NFP) + 2 * f]     = c0;
        wcp[pair * (2 * NFP) + 2 * f + 1] = c1;
        wsp[pair * (2 * NFP) + 2 * f]     = s0;
        wsp[pair * (2 * NFP) + 2 * f + 1] = s1;
    }
    if (idx < NF * NK) {
        int f = idx / NK, k = idx % NK;
        float w = (f == 0 || f == NF - 1) ? (1.0f / 256.0f) : (2.0f / 256.0f);
        tir[idx] = w * __cosf(TWO_PI * (float)f * (float)(127 - k) / 256.0f);
    }
}

// ---------------- K1: per-(b,c) mean ----------------
__global__ void __launch_bounds__(256) k_mean(const float* __restrict__ x,
                                              float* __restrict__ means) {
    int bc = blockIdx.x;
    const float* xr = x + (size_t)bc * NT;
    float s = 0.f;
    for (int t = threadIdx.x; t < NT; t += 256) s += xr[t];
    __shared__ float red[256];
    red[threadIdx.x] = s; __syncthreads();
    for (int st = 128; st > 0; st >>= 1) {
        if (threadIdx.x < st) red[threadIdx.x] += red[threadIdx.x + st];
        __syncthreads();
    }
    if (threadIdx.x == 0) means[bc] = red[0] / (float)NT;
}

// ---------------- K2: Welch PSD as fp32 WMMA GEMM ----------------
// Segment tile staged by the Tensor Data Mover: 256x48 tile, row stride 128
// (50% overlap), tensor_dim1=45 zero-fills pad rows 45..47 via OOB semantics,
// TDM pad feature inserts 4 dwords per 256 -> LDS row stride 260 (bank-safe).
#define SEG_STRIDE 260
__global__ void __launch_bounds__(288) k_psd(const float* __restrict__ x,
                                             const float* __restrict__ wcp,
                                             const float* __restrict__ wsp,
                                             float* __restrict__ psd) {
    int bc   = blockIdx.x;
    int ft   = threadIdx.x >> 5;     // 9 waves: freq tile
    int lane = threadIdx.x & 31;
    int nn   = lane & 15;            // A row (segment) / B col (freq)
    int hi   = lane >> 4;            // K-pair select (0,1) vs (2,3)
    int f    = ft * 16 + nn;         // 0..143; table zero-padded beyond 128
    const float* xr = x + (size_t)bc * NT;

    __shared__ float segl[48 * SEG_STRIDE];
    if (threadIdx.x < 32) {          // wave 0 drives the DMA, others park at barrier
        tdm_load_tile((unsigned)(size_t)(void*)segl,
                      (unsigned long long)(size_t)xr,
                      /*tensor_dim0=*/256, /*tensor_dim1=*/NSEG,
                      /*tile0=*/256, /*tile1=*/48, /*stride0=*/128,
                      /*pad_interval=*/7, /*pad_amount=*/3, /*pad_en=*/true);
        __builtin_amdgcn_s_wait_tensorcnt(0);
    }
    __syncthreads();

    v8f accRe0 = {}, accRe1 = {}, accRe2 = {};
    v8f accIm0 = {}, accIm1 = {}, accIm2 = {};
#pragma unroll 2
    for (int k4 = 0; k4 < 64; ++k4) {            // K = 256 in steps of 4
        int n0 = 4 * k4 + 2 * hi;                // even -> 8B-aligned loads
        v2f a0 = *(const v2f*)&segl[(nn)      * SEG_STRIDE + n0];
        v2f a1 = *(const v2f*)&segl[(nn + 16) * SEG_STRIDE + n0];
        v2f a2 = *(const v2f*)&segl[(nn + 32) * SEG_STRIDE + n0];
        v2f bre = *(const v2f*)&wcp[(n0 >> 1) * (2 * NFP) + 2 * f];
        v2f bim = *(const v2f*)&wsp[(n0 >> 1) * (2 * NFP) + 2 * f];
        accRe0 = wmma4(a0, bre, accRe0);
        accIm0 = wmma4(a0, bim, accIm0);
        accRe1 = wmma4(a1, bre, accRe1);
        accIm1 = wmma4(a1, bim, accIm1);
        accRe2 = wmma4(a2, bre, accRe2);
        accIm2 = wmma4(a2, bim, accIm2);
    }
    float psum = 0.f;
    for (int j = 0; j < 8; ++j)
        psum += accRe0[j] * accRe0[j] + accIm0[j] * accIm0[j]
              + accRe1[j] * accRe1[j] + accIm1[j] * accIm1[j]
              + accRe2[j] * accRe2[j] + accIm2[j] * accIm2[j];
    psum += __shfl_down(psum, 16, 32);           // fold M=8..15 half
    if (hi == 0 && f < NF) {
        float scale = (f == 0) ? 0.f : ((f == NF - 1) ? 1.f : 2.f);
        psd[(size_t)bc * NF + f] = psum * scale / (256.0f * (float)NSEG);
    }
}

// ---------------- K3: barycenter over batch ----------------
__global__ void __launch_bounds__(256) k_bary(const float* __restrict__ psd,
                                              float* __restrict__ bary) {
    int cf = blockIdx.x;                          // c*129+f, 1032 total
    float s = 0.f;
    for (int b = threadIdx.x; b < NB; b += 256)
        s += sqrtf(psd[(size_t)b * (NC * NF) + cf] + 1e-7f);
    __shared__ float red[256];
    red[threadIdx.x] = s; __syncthreads();
    for (int st = 128; st > 0; st >>= 1) {
        if (threadIdx.x < st) red[threadIdx.x] += red[threadIdx.x + st];
        __syncthreads();
    }
    if (threadIdx.x == 0) { float v = red[0] / (float)NF; bary[cf] = v * v; }
}

// ---------------- K4: whitening filter (irfft+shift+flip fused) ----------------
__global__ void __launch_bounds__(256) k_filt(const float* __restrict__ psd,
                                              const float* __restrict__ bary,
                                              const float* __restrict__ tir,
                                              float* __restrict__ hbuf) {
    int bc = blockIdx.x, c = bc & (NC - 1);
    __shared__ float Dc[NF];
    if (threadIdx.x < NF) {
        float p = psd[(size_t)bc * NF + threadIdx.x] + 1e-7f;
        Dc[threadIdx.x] = sqrtf(bary[c * NF + threadIdx.x]) / sqrtf(p);
    }
    __syncthreads();
    int k = threadIdx.x;
    float h = 0.f;
    for (int f = 0; f < NF; ++f) h += Dc[f] * tir[f * NK + k];
    hbuf[(size_t)bc * NK + k] = h;
}

// ---------------- K5: depthwise conv as Toeplitz fp32 WMMA GEMM ----------------
// out[t0+16p+m] = sum_kk x_pad[t0+16p+kk-127] * h[kk-m], kk in [0,272)
// XSK skew: 2 dwords per 64-dword line -> 16 lanes hit 16 distinct banks while
// keeping 8B pair contiguity/alignment (indices always even).
#define XSK(i) ((i) + 2 * ((i) >> 6))
__global__ void __launch_bounds__(256) k_conv(const float* __restrict__ x,
                                              const float* __restrict__ means,
                                              const float* __restrict__ hbuf,
                                              float* __restrict__ out) {
    int blk = blockIdx.x;
    int bc = blk / 3, chunk = blk % 3;            // 3 chunks of 2048 outputs
    int wave = threadIdx.x >> 5, lane = threadIdx.x & 31;
    int nn = lane & 15, hi = lane >> 4;

    __shared__ float xl[2400];                    // skewed padded signal window
    __shared__ float hsp[288];                    // h with 16 zeros front/back
    __shared__ float ot[2112];                    // skewed output transpose stage
    const float* xr = x + (size_t)bc * NT;
    float mu = means[bc];                         // mean folded into staging (TDM can't)
    int base = chunk * 2048 - 127;                // 'same' padding: left 127
    for (int i = threadIdx.x; i < 2319; i += 256) {
        int t = base + i;
        xl[XSK(i)] = (t >= 0 && t < NT) ? (xr[t] - mu) : 0.f;
    }
    hsp[16 + threadIdx.x] = hbuf[(size_t)bc * NK + threadIdx.x];
    if (threadIdx.x < 16)   hsp[threadIdx.x] = 0.f;
    if (threadIdx.x >= 240) hsp[threadIdx.x + 32] = 0.f;
    __syncthreads();

    int tile = wave * 256;                        // each wave: 256 outputs
    v8f acc = {};
#pragma unroll 2
    for (int k4 = 0; k4 < 68; ++k4) {             // K = 272 in steps of 4
        int kk = 4 * k4 + 2 * hi;                 // even -> aligned A b64
        v2f a = *(const v2f*)&xl[XSK(tile + 16 * nn + kk)];
        v2f b = { hsp[16 + kk - nn], hsp[17 + kk - nn] };
        acc = wmma4(a, b, acc);
    }
    // transpose through LDS -> two coalesced b128 stores per thread
    for (int j = 0; j < 8; ++j)
        ot[XSK(wave * 256 + 16 * (j + 8 * hi) + nn)] = acc[j];
    __syncthreads();
    int t0 = chunk * 2048 + threadIdx.x * 8;      // chunk length 1904 = 238*8: clean tail
    if (t0 + 7 < NT) {
        v2f p0 = *(const v2f*)&ot[XSK(threadIdx.x * 8 + 0)];
        v2f p1 = *(const v2f*)&ot[XSK(threadIdx.x * 8 + 2)];
        v2f p2 = *(const v2f*)&ot[XSK(threadIdx.x * 8 + 4)];
        v2f p3 = *(const v2f*)&ot[XSK(threadIdx.x * 8 + 6)];
        float4 q0 = {p0.x, p0.y, p1.x, p1.y};
        float4 q1 = {p2.x, p2.y, p3.x, p3.y};
        *(float4*)&out[(size_t)bc * NT + t0]     = q0;
        *(float4*)&out[(size_t)bc * NT + t0 + 4] = q1;
    }
}

// ---------------- launch ----------------
extern "C" void kernel_launch(void* const* d_in, const int* in_sizes, int n_in,
                              void* d_out, int out_size, void* d_ws, size_t ws_size,
                              hipStream_t stream) {
    (void)in_sizes; (void)n_in; (void)out_size; (void)ws_size;
    const float* x = (const float*)d_in[0];
    float* out = (float*)d_out;

    float* ws    = (float*)d_ws;
    float* means = ws;                          // 4096
    float* psd   = means + NBC;                 // 4096*129  = 528384
    float* bary  = psd + (size_t)NBC * NF;      // 1032
    float* hbuf  = bary + NC * NF;              // 4096*256  = 1048576
    float* wcp   = hbuf + (size_t)NBC * NK;     // 128*144*2 = 36864
    float* wsp   = wcp + 128 * 2 * NFP;         // 36864
    float* tir   = wsp + 128 * 2 * NFP;         // 33024   (~6.8 MB total)

    k_init<<<(NF * NK + 255) / 256, 256, 0, stream>>>(wcp, wsp, tir);
    k_mean<<<NBC, 256, 0, stream>>>(x, means);
    k_psd <<<NBC, 288, 0, stream>>>(x, wcp, wsp, psd);
    k_bary<<<NC * NF, 256, 0, stream>>>(psd, bary);
    k_filt<<<NBC, 256, 0, stream>>>(psd, bary, tir, hbuf);
    k_conv<<<NBC * 3, 256, 0, stream>>>(x, means, hbuf, out);
}